// FSDDRC_66400194396290
// MI455X (gfx1250) — compile-verified
//
#include <hip/hip_runtime.h>

// CDNA5 (gfx1250) reservoir-computing scan.
//   f_t = clip(4*(A f_{t-1} + x_t), -1, 1);  out_t = [f_t, reluleak128(f_t)]
// Bandwidth-bound (384MB traffic, <1 GFLOP). One wave32 owns 16 batch columns;
// the 8x8 matvec runs on V_WMMA_F32_16X16X4_F32 (two chained K=4 WMMAs), with
// the weight rows doubled so the upper half of D is the duplicated
// pre-activation used for the reluleak output AND for a shuffle-free state
// update (pure v_cndmask lane selects, no DS/permute traffic).

typedef __attribute__((ext_vector_type(2))) float v2f;
typedef __attribute__((ext_vector_type(4))) float v4f;
typedef __attribute__((ext_vector_type(8))) float v8f;

#define RC_T 1024
#define RC_N 8

__global__ __launch_bounds__(32)
void FSDDRC_rc_wmma_kernel(const float* __restrict__ x,
                           const float* __restrict__ A,
                           float* __restrict__ out) {
  const int lane = threadIdx.x & 31;
  const int nl   = lane & 15;        // N column of the WMMA tile = batch-in-tile
  const bool hi  = lane >= 16;       // upper lane half handles M = 8..15 (dup rows)
  const int b    = blockIdx.x * 16 + nl;   // this lane's batch element

  // ---- A-operand (constant, 16x4 per WMMA): Wmat[M][K] = A[M%8][K] ----
  // Layout: lanes 0-15 hold M=lane, K={0,1} (vgpr0,1); lanes 16-31 hold K={2,3}.
  const int ar = (nl & 7) * 8;
  const int k0 = hi ? 2 : 0;         // first WMMA covers K=0..3
  const int k4 = hi ? 6 : 4;         // second WMMA covers K=4..7
  v2f wa01, wa23;
  wa01.x = A[ar + k0];
  wa01.y = A[ar + k0 + 1];
  wa23.x = A[ar + k4];
  wa23.y = A[ar + k4 + 1];

  // ---- B-operand = state F[K][N=batch]; K split like A's layout. f0 = 0 ----
  v2f fb01 = {0.0f, 0.0f};           // K = hi ? {2,3} : {0,1}
  v2f fb45 = {0.0f, 0.0f};           // K = hi ? {6,7} : {4,5}

  // x:  [B, 8, T]  -> lane base: this batch's 8xT slab (contiguous in t)
  // out:[B, 16, T] -> low lanes write comps 0..7 (f), high lanes 8..15 (leak)
  const float* xb = x   + (size_t)b * (RC_N * RC_T);
  float*       ob = out + (size_t)b * (16 * RC_T) + (hi ? 8 * RC_T : 0);

  // ---- software-pipelined x tile (4 time steps, b128 per component) ----
  v4f xv[8];
#pragma unroll
  for (int v = 0; v < 8; ++v)
    xv[v] = *(const v4f*)(xb + v * RC_T);

  for (int t0 = 0; t0 < RC_T; t0 += 4) {
    v4f xn[8];
    const bool more = (t0 + 4) < RC_T;
    if (more) {
#pragma unroll
      for (int v = 0; v < 8; ++v)
        xn[v] = *(const v4f*)(xb + v * RC_T + (t0 + 4));
    }

    v4f ov[8];                       // 4 t-steps of output per D-vgpr
#pragma unroll
    for (int s = 0; s < 4; ++s) {
      // C-operand: C[M][N] = x[b, M%8, t]; comp index == vgpr index for both halves
      v8f c;
#pragma unroll
      for (int v = 0; v < 8; ++v) c[v] = xv[v][s];

      // pre = Wmat(16x8) x F(8x16) + X  via two chained K=4 WMMAs
      v8f d = __builtin_amdgcn_wmma_f32_16x16x4_f32(false, wa01, false, fb01,
                                                    (short)0, c, false, false);
      d     = __builtin_amdgcn_wmma_f32_16x16x4_f32(false, wa23, false, fb45,
                                                    (short)0, d, false, false);

      float fn[8];
#pragma unroll
      for (int v = 0; v < 8; ++v) {
        float p = 4.0f * d[v];
        p = fminf(1.0f, fmaxf(-1.0f, p));          // nonlin = clip(4x,-1,1)
        fn[v] = p;                                  // f_new (dup rows -> same value)
        ov[v][s] = hi ? (p >= 0.0f ? p : p * 0.0078125f)  // reluleak128 half
                      : p;                                 // f half
      }
      // State update, shuffle-free: the duplicated M rows put f_new[2,3,6,7]
      // on the high lanes already (D[v] high half = pre[comp v] too).
      fb01.x = hi ? fn[2] : fn[0];
      fb01.y = hi ? fn[3] : fn[1];
      fb45.x = hi ? fn[6] : fn[4];
      fb45.y = hi ? fn[7] : fn[5];
    }

    // out is write-once and larger than L2 -> non-temporal b128 stores
#pragma unroll
    for (int v = 0; v < 8; ++v)
      __builtin_nontemporal_store(ov[v], (v4f*)(ob + v * RC_T + t0));

    if (more) {
#pragma unroll
      for (int v = 0; v < 8; ++v) xv[v] = xn[v];
    }
  }
}

extern "C" void kernel_launch(void* const* d_in, const int* in_sizes, int n_in,
                              void* d_out, int out_size, void* d_ws, size_t ws_size,
                              hipStream_t stream) {
  const float* x = (const float*)d_in[0];   // [B, 8, 1024]
  const float* A = (const float*)d_in[1];   // [8, 8]
  float* out = (float*)d_out;               // [B, 16, 1024]

  const int B = in_sizes[0] / (RC_N * RC_T);   // 4096
  const int tiles = B / 16;                    // 16 batch columns per wave
  // One wave32 per workgroup, one 16-batch WMMA tile per wave; 256 blocks
  // spread across all WGPs for bandwidth.
  FSDDRC_rc_wmma_kernel<<<tiles, 32, 0, stream>>>(x, A, out);
}